// MoAGate_32134945309383
// MI455X (gfx1250) — compile-verified
//
#include <hip/hip_runtime.h>

typedef float v2f __attribute__((ext_vector_type(2)));
typedef float v8f __attribute__((ext_vector_type(8)));

#define D_DIM      1024
#define LDS_ROWS   10        // 9 weight rows + 1 shared zero row (for M=9..15)
#define ROWSTRIDE  1028      // pad: stride % 64 banks == 4 -> conflict-free A fetch
#define NT         4         // 16-token tiles per wave (amortizes A-operand LDS load)
#define WAVES_PER_BLK 4
#define THREADS    (WAVES_PER_BLK * 32)
#define TOK_PER_WAVE (NT * 16)
#define TOK_PER_BLK  (WAVES_PER_BLK * TOK_PER_WAVE)

__global__ __launch_bounds__(THREADS)
void moa_gate_kernel(const float* __restrict__ tokens,
                     const float* __restrict__ Wq,
                     const float* __restrict__ Wk,
                     const float* __restrict__ Wv,
                     float* __restrict__ out)
{
    __shared__ float w[LDS_ROWS * ROWSTRIDE];   // ~41 KB

    const int tid = threadIdx.x;

    // Cooperative stage of combined weight matrix: rows 0-2=Wq, 3-5=Wk, 6-8=Wv, row 9=zeros.
    for (int i = tid; i < LDS_ROWS * D_DIM; i += THREADS) {
        int r = i >> 10;
        int c = i & (D_DIM - 1);
        float val = 0.0f;
        if      (r < 3) val = Wq[r * D_DIM + c];
        else if (r < 6) val = Wk[(r - 3) * D_DIM + c];
        else if (r < 9) val = Wv[(r - 6) * D_DIM + c];
        w[r * ROWSTRIDE + c] = val;
    }
    __syncthreads();

    const int lane = tid & 31;
    const int wave = tid >> 5;
    const int l15  = lane & 15;
    const int hi2  = (lane >> 4) << 1;          // 0 for lanes 0-15, 2 for lanes 16-31

    // A-operand (weights) LDS address: row M = l15 (M>=9 -> shared zero row), K pair per half-wave.
    const int wRow = (l15 < 9) ? l15 : 9;
    const int wOff = wRow * ROWSTRIDE + hi2;

    const int waveTokenBase = blockIdx.x * TOK_PER_BLK + wave * TOK_PER_WAVE;

    // B-operand (tokens) per-lane row pointers: lane holds token (tile*16 + l15),
    // reading K pair (k+hi2, k+hi2+1) each step -> global_load_b64, WMMA-native layout.
    const float* p[NT];
    #pragma unroll
    for (int t = 0; t < NT; ++t)
        p[t] = tokens + (size_t)(waveTokenBase + t * 16 + l15) * D_DIM + hi2;

    v8f acc[NT] = {};   // D = [16 outputs x 16 tokens] per tile

    #pragma unroll 4
    for (int k = 0; k < D_DIM; k += 4) {
        v2f a = *(const v2f*)(&w[wOff + k]);          // ds_load_b64 (shared across NT tiles)
        v2f b[NT];
        #pragma unroll
        for (int t = 0; t < NT; ++t)
            b[t] = *(const v2f*)(p[t] + k);           // global_load_b64
        #pragma unroll
        for (int t = 0; t < NT; ++t)
            acc[t] = __builtin_amdgcn_wmma_f32_16x16x4_f32(
                         false, a, false, b[t], (short)0, acc[t], false, false);
    }

    // v2 (output row M=8) for token N sits in VGPR0 of lane N+16.
    float v2x[NT];
    #pragma unroll
    for (int t = 0; t < NT; ++t)
        v2x[t] = __shfl(acc[t][0], l15 + 16, 32);

    if (lane < 16) {
        #pragma unroll
        for (int t = 0; t < NT; ++t) {
            const float qe[3] = { acc[t][0], acc[t][1], acc[t][2] };
            const float k0 = acc[t][3], k1 = acc[t][4], k2 = acc[t][5];
            const float vv0 = acc[t][6], vv1 = acc[t][7], vv2 = v2x[t];

            float lg[3];
            #pragma unroll
            for (int e = 0; e < 3; ++e) {
                float x0 = qe[e] * k0, x1 = qe[e] * k1, x2 = qe[e] * k2;
                float m  = fmaxf(x0, fmaxf(x1, x2));
                float p0 = __expf(x0 - m), p1 = __expf(x1 - m), p2 = __expf(x2 - m);
                float s  = p0 + p1 + p2;
                lg[e] = (p0 * vv0 + p1 * vv1 + p2 * vv2) / s;
            }
            float m2 = fmaxf(lg[0], fmaxf(lg[1], lg[2]));
            float e0 = __expf(lg[0] - m2), e1 = __expf(lg[1] - m2), e2 = __expf(lg[2] - m2);
            float inv = 1.0f / (e0 + e1 + e2);

            float* o = out + (size_t)(waveTokenBase + t * 16 + l15) * 3;
            o[0] = e0 * inv;
            o[1] = e1 * inv;
            o[2] = e2 * inv;
        }
    }
}

extern "C" void kernel_launch(void* const* d_in, const int* in_sizes, int n_in,
                              void* d_out, int out_size, void* d_ws, size_t ws_size,
                              hipStream_t stream) {
    (void)in_sizes; (void)n_in; (void)d_ws; (void)ws_size; (void)out_size;
    const float* tokens = (const float*)d_in[0];
    const float* Wq     = (const float*)d_in[1];
    const float* Wk     = (const float*)d_in[2];
    const float* Wv     = (const float*)d_in[3];
    float*       out    = (float*)d_out;

    const int total_tokens = 8 * 4096;                 // B * S
    const int blocks = total_tokens / TOK_PER_BLK;     // 128
    moa_gate_kernel<<<blocks, THREADS, 0, stream>>>(tokens, Wq, Wk, Wv, out);
}